// AlpacaMoca_15109694948019
// MI455X (gfx1250) — compile-verified
//
#include <hip/hip_runtime.h>

typedef __attribute__((ext_vector_type(2))) float v2f;
typedef __attribute__((ext_vector_type(4))) float v4f;
typedef __attribute__((ext_vector_type(8))) float v8f;

#define B_ 16
#define Z_ 64
#define U_ 8
#define P_ 128
#define X_ 64
#define H_ 128

// ---------------------------------------------------------------------------
// Kernel 1: encoder MLP phi = (elu chain)(x) using V_WMMA_F32_16X16X4_F32.
// One block, 256 threads = 8 waves; wave w computes output columns
// [16w, 16w+16) of every layer. Activations staged in LDS.
// A-frag layout (16x4 f32): lanes 0-15 hold row m, K = k0+0 / k0+1 in v0/v1;
// lanes 16-31 hold row m, K = k0+2 / k0+3.
// B-frag layout (4x16 f32): mirrored (K across lane-halves/VGPRs, N = lane%16).
// C/D layout (16x16 f32): VGPR r, lane<16 -> M=r, lane>=16 -> M=r+8, N=lane%16.
// ---------------------------------------------------------------------------
__device__ __forceinline__ v8f wmma_tile_f32(const float* __restrict__ actLds, int ldA,
                                             const float* __restrict__ W, int ldW,
                                             int K, int nBase, int lane) {
    v8f c = {};
    const int m    = lane & 15;
    const int kSel = (lane < 16) ? 0 : 2;
    const int n    = nBase + (lane & 15);
    for (int k0 = 0; k0 < K; k0 += 4) {
        v2f a, b;
        a.x = actLds[m * ldA + k0 + kSel + 0];
        a.y = actLds[m * ldA + k0 + kSel + 1];
        b.x = W[(k0 + kSel + 0) * ldW + n];
        b.y = W[(k0 + kSel + 1) * ldW + n];
        // 8 args: (neg_a, A, neg_b, B, c_mod, C, reuse_a, reuse_b)
        c = __builtin_amdgcn_wmma_f32_16x16x4_f32(false, a, false, b,
                                                  (short)0, c, false, false);
    }
    return c;
}

__device__ __forceinline__ void store_tile(float* __restrict__ dst, int ldD,
                                           const v8f c, const float* __restrict__ bias,
                                           int nBase, int lane, bool doElu) {
    const int n    = nBase + (lane & 15);
    const int mOff = (lane < 16) ? 0 : 8;
    const float bv = bias[n];
#pragma unroll
    for (int r = 0; r < 8; ++r) {
        float v = c[r] + bv;
        if (doElu) v = (v > 0.0f) ? v : (expf(v) - 1.0f);
        dst[(r + mOff) * ldD + n] = v;
    }
}

__global__ __launch_bounds__(256) void encoder_kernel(
    const float* __restrict__ x,
    const float* __restrict__ W1, const float* __restrict__ b1,
    const float* __restrict__ W2, const float* __restrict__ b2,
    const float* __restrict__ W3, const float* __restrict__ b3,
    const float* __restrict__ W4, const float* __restrict__ b4,
    float* __restrict__ phi_out) {
    __shared__ __align__(16) float bufA[B_ * H_];
    __shared__ __align__(16) float bufB[B_ * H_];

    const int t    = threadIdx.x;
    const int lane = t & 31;
    const int wave = t >> 5;
    const int nBase = wave * 16;

    // stage x (16x64) into bufA, stride X_
    for (int i = t; i < B_ * X_; i += 256) bufA[i] = x[i];
    __syncthreads();

    // layer 1: (16x64)@(64x128)+b1, ELU -> bufB
    {
        v8f c = wmma_tile_f32(bufA, X_, W1, H_, X_, nBase, lane);
        store_tile(bufB, H_, c, b1, nBase, lane, true);
    }
    __syncthreads();
    // layer 2: bufB -> bufA
    {
        v8f c = wmma_tile_f32(bufB, H_, W2, H_, H_, nBase, lane);
        store_tile(bufA, H_, c, b2, nBase, lane, true);
    }
    __syncthreads();
    // layer 3: bufA -> bufB
    {
        v8f c = wmma_tile_f32(bufA, H_, W3, H_, H_, nBase, lane);
        store_tile(bufB, H_, c, b3, nBase, lane, true);
    }
    __syncthreads();
    // layer 4 (no ELU): bufB -> phi_out (global, 16x128)
    {
        v8f c = wmma_tile_f32(bufB, H_, W4, P_, H_, nBase, lane);
        const int n    = nBase + (lane & 15);
        const int mOff = (lane < 16) ? 0 : 8;
        const float bv = b4[n];
#pragma unroll
        for (int r = 0; r < 8; ++r) {
            phi_out[(r + mOff) * P_ + n] = c[r] + bv;
        }
    }
}

// ---------------------------------------------------------------------------
// Kernel 2: streaming bilinear forms. One block per (b,z,u); each block
// streams its 64 KB tile of Linv exactly once (non-temporal b128 loads),
// computing  mu = phi^T L q   and   s = phi^T L phi  in the same pass.
// 256 threads: thread t owns row q = t/2, columns [64*(t&1), 64*(t&1)+64).
// ---------------------------------------------------------------------------
__global__ __launch_bounds__(256) void bilinear_kernel(
    const float* __restrict__ Linv, const float* __restrict__ Q,
    const float* __restrict__ phi, const float* __restrict__ logSigEps,
    float* __restrict__ mu_out, float* __restrict__ cov_out) {
    __shared__ __align__(16) float s_phi[P_];
    __shared__ __align__(16) float s_q[P_];
    __shared__ float red_mu[8];
    __shared__ float red_s[8];

    const int task = blockIdx.x;              // b*Z*U + z*U + u
    const int b    = task >> 9;               // / (Z_*U_) = 512
    const int u    = task & (U_ - 1);

    const int t = threadIdx.x;
    if (t < P_) s_phi[t] = phi[b * P_ + t];
    else        s_q[t - P_] = Q[(size_t)task * P_ + (t - P_)];
    __syncthreads();

    const int q  = t >> 1;
    const int p0 = (t & 1) << 6;              // 0 or 64
    const v4f* __restrict__ L4 =
        (const v4f*)(Linv + (size_t)task * (P_ * P_) + (size_t)q * P_ + p0);
    const v4f* __restrict__ q4 = (const v4f*)(s_q + p0);
    const v4f* __restrict__ f4 = (const v4f*)(s_phi + p0);

    float acc1 = 0.0f, acc2 = 0.0f;
#pragma unroll
    for (int i = 0; i < 16; ++i) {
        const v4f L  = __builtin_nontemporal_load(&L4[i]);   // stream: 512MB, no reuse
        const v4f qq = q4[i];
        const v4f ff = f4[i];
        acc1 = fmaf(L.x, qq.x, acc1);
        acc1 = fmaf(L.y, qq.y, acc1);
        acc1 = fmaf(L.z, qq.z, acc1);
        acc1 = fmaf(L.w, qq.w, acc1);
        acc2 = fmaf(L.x, ff.x, acc2);
        acc2 = fmaf(L.y, ff.y, acc2);
        acc2 = fmaf(L.z, ff.z, acc2);
        acc2 = fmaf(L.w, ff.w, acc2);
    }
    const float phiq = s_phi[q];
    float vmu = phiq * acc1;
    float vs  = phiq * acc2;

    // wave32 tree reduction
#pragma unroll
    for (int off = 16; off > 0; off >>= 1) {
        vmu += __shfl_down(vmu, off, 32);
        vs  += __shfl_down(vs,  off, 32);
    }
    const int lane = t & 31;
    const int wave = t >> 5;
    if (lane == 0) { red_mu[wave] = vmu; red_s[wave] = vs; }
    __syncthreads();

    if (t == 0) {
        float m = 0.0f, s = 0.0f;
#pragma unroll
        for (int w = 0; w < 8; ++w) { m += red_mu[w]; s += red_s[w]; }
        mu_out[task]  = m;
        cov_out[task] = expf(logSigEps[u]) * (1.0f + s);
    }
}

// ---------------------------------------------------------------------------
// Launch. Inputs (setup_inputs order):
// 0:x 1:Linv 2:Q 3:W1 4:b1 5:W2 6:b2 7:W3 8:b3 9:W4 10:b4 11:logSigEps
// Output: mu (8192) ++ pred_cov_diag (8192), fp32.
// ---------------------------------------------------------------------------
extern "C" void kernel_launch(void* const* d_in, const int* in_sizes, int n_in,
                              void* d_out, int out_size, void* d_ws, size_t ws_size,
                              hipStream_t stream) {
    const float* x      = (const float*)d_in[0];
    const float* Linv   = (const float*)d_in[1];
    const float* Q      = (const float*)d_in[2];
    const float* W1     = (const float*)d_in[3];
    const float* b1     = (const float*)d_in[4];
    const float* W2     = (const float*)d_in[5];
    const float* b2     = (const float*)d_in[6];
    const float* W3     = (const float*)d_in[7];
    const float* b3     = (const float*)d_in[8];
    const float* W4     = (const float*)d_in[9];
    const float* b4     = (const float*)d_in[10];
    const float* logSig = (const float*)d_in[11];

    float* phi = (float*)d_ws;                 // 16*128 floats
    float* out = (float*)d_out;
    float* mu_out  = out;                      // first 8192
    float* cov_out = out + B_ * Z_ * U_;       // next 8192

    encoder_kernel<<<1, 256, 0, stream>>>(x, W1, b1, W2, b2, W3, b3, W4, b4, phi);
    bilinear_kernel<<<B_ * Z_ * U_, 256, 0, stream>>>(Linv, Q, phi, logSig,
                                                      mu_out, cov_out);
}